// CRF_57655640981618
// MI455X (gfx1250) — compile-verified
//
#include <hip/hip_runtime.h>
#include <hip/hip_bf16.h>

typedef __attribute__((ext_vector_type(2))) float v2f;
typedef __attribute__((ext_vector_type(8))) float v8f;

#define CRF_B 256
#define CRF_S 512
#define CRF_T 64

#define LOG2E 1.4426950408889634f
#define LN2   0.6931471805599453f

// ---------------------------------------------------------------------------
// Numerator: per-batch gather score. One thread per batch.
// ---------------------------------------------------------------------------
__global__ void crf_num_kernel(const float* __restrict__ em,
                               const long long* __restrict__ tags,
                               const int* __restrict__ mask,
                               const float* __restrict__ startT,
                               const float* __restrict__ endT,
                               const float* __restrict__ trans,
                               float* __restrict__ num) {
    int b = blockIdx.x * blockDim.x + threadIdx.x;
    if (b >= CRF_B) return;
    const long long* tg = tags + (size_t)b * CRF_S;
    const int*       mk = mask + (size_t)b * CRF_S;
    const float*     e  = em   + (size_t)b * CRF_S * CRF_T;
    int tprev = (int)tg[0];
    float score = startT[tprev] + e[tprev];
    int last = tprev;
    for (int s = 1; s < CRF_S; ++s) {
        int ts = (int)tg[s];
        float add = trans[tprev * CRF_T + ts] + e[(size_t)s * CRF_T + ts];
        if (mk[s]) { score += add; last = ts; }
        tprev = ts;
    }
    score += endT[last];
    num[b] = score;
}

// ---------------------------------------------------------------------------
// Denominator: forward algorithm as exp-domain GEMM recurrence, base-2 log
// domain so exp/log lower to raw v_exp_f32 / v_log_f32 (no libm denorm
// guards — the WMMA accumulator is provably >= 0.9).
//   One workgroup = 16 batch rows, 4 waves, one wave per 16-state tile.
//   Scores live in WMMA C/D accumulator layout; E = exp(transitions) lives
//   in per-wave B fragments (loaded once). Mask tile preloaded to LDS.
//   3 barriers/step (the 4th is provably redundant: partmax WAR is covered
//   by barriers B+C, rowmaxC WAR by next iteration's barrier A, P WAR by
//   barriers A+B).
// ---------------------------------------------------------------------------
__global__ __launch_bounds__(128) void crf_fwd_kernel(
        const float* __restrict__ em,
        const int*   __restrict__ mask,
        const float* __restrict__ startT,
        const float* __restrict__ endT,
        const float* __restrict__ trans,
        float* __restrict__ denom) {
    __shared__ float P[16][68];          // p = exp2(c2 - m2), padded stride
    __shared__ float partmax[4][16];     // per-wave row maxes
    __shared__ float rowmaxC[16];        // combined row max (base-2 domain)
    __shared__ float partsum[4][16];     // final lse partial sums
    __shared__ int   mlds[16][CRF_S + 1];// mask tile, padded row stride

    const int tid   = threadIdx.x;
    const int lane  = tid & 31;
    const int wave  = tid >> 5;
    const int nl    = lane & 15;              // column within 16-wide tile
    const int hi    = (lane >> 4) & 1;        // half-wave select
    const int bbase = blockIdx.x * 16;        // batch tile base
    const int nbase = wave * 16;              // state tile base

    // --- Preload the 16 x 512 mask tile into LDS (coalesced, one time).
    for (int i = tid; i < 16 * CRF_S; i += 128) {
        int r = i >> 9;            // i / 512
        int scol = i & (CRF_S - 1);
        mlds[r][scol] = mask[(size_t)(bbase + r) * CRF_S + scol];
    }

    // --- Load constant B fragments: E = exp(trans) = exp2(trans*log2e).
    // B (4x16) layout: VGPR0 = rows K=0 (lanes 0-15) / K=2 (lanes 16-31),
    //                  VGPR1 = rows K=1 / K=3.
    v2f bfrag[16];
#pragma unroll
    for (int kk = 0; kk < 16; ++kk) {
        int k0 = 4 * kk + (hi ? 2 : 0);
        bfrag[kk][0] = __builtin_amdgcn_exp2f(trans[(size_t)k0 * CRF_T + nbase + nl] * LOG2E);
        bfrag[kk][1] = __builtin_amdgcn_exp2f(trans[(size_t)(k0 + 1) * CRF_T + nbase + nl] * LOG2E);
    }

    // --- Per-row emission base pointers (row M = v + 8*hi).
    const float* emrow[8];
#pragma unroll
    for (int v = 0; v < 8; ++v) {
        int M = v + hi * 8;
        emrow[v] = em + ((size_t)(bbase + M) * CRF_S) * CRF_T + nbase + nl;
    }

    // --- Init (base-2 domain): c2 = (start + em[0]) * log2e.
    // C/D layout: VGPR v = row M=v (lanes 0-15) / M=v+8 (lanes 16-31),
    // N = lane&15.
    const float st2 = startT[nbase + nl] * LOG2E;
    v8f c;
#pragma unroll
    for (int v = 0; v < 8; ++v) c[v] = fmaf(emrow[v][0], LOG2E, st2);

    __syncthreads();   // mask tile visible before the scan

    for (int s = 1; s < CRF_S; ++s) {
        // (1) per-wave row max via 16-lane butterfly
#pragma unroll
        for (int v = 0; v < 8; ++v) {
            float x = c[v];
            x = fmaxf(x, __shfl_xor(x, 1, 32));
            x = fmaxf(x, __shfl_xor(x, 2, 32));
            x = fmaxf(x, __shfl_xor(x, 4, 32));
            x = fmaxf(x, __shfl_xor(x, 8, 32));
            if (nl == 0) partmax[wave][v + hi * 8] = x;   // lanes 0 and 16
        }
        __syncthreads();   // barrier A
        // (2) combine row maxes across waves
        if (tid < 16) {
            rowmaxC[tid] = fmaxf(fmaxf(partmax[0][tid], partmax[1][tid]),
                                 fmaxf(partmax[2][tid], partmax[3][tid]));
        }
        __syncthreads();   // barrier B
        // (3) p = exp2(c2 - m2) -> LDS  (raw v_exp_f32)
#pragma unroll
        for (int v = 0; v < 8; ++v) {
            int M = v + hi * 8;
            P[M][nbase + nl] = __builtin_amdgcn_exp2f(c[v] - rowmaxC[M]);
        }
        __syncthreads();   // barrier C
        // (4) acc = P(16x64) x E(64x16) via 16 chained f32 WMMAs.
        // A (16x4) layout: M = lane&15; VGPR0 = K {0|2}, VGPR1 = K {1|3}.
        v8f acc;
#pragma unroll
        for (int v = 0; v < 8; ++v) acc[v] = 0.0f;
#pragma unroll
        for (int kk = 0; kk < 16; ++kk) {
            int kb = 4 * kk + (hi ? 2 : 0);
            v2f a;
            a[0] = P[nl][kb];
            a[1] = P[nl][kb + 1];
            acc = __builtin_amdgcn_wmma_f32_16x16x4_f32(
                false, a, false, bfrag[kk], (short)0, acc, false, false);
        }
        // (5) c2' = em*log2e + m2 + log2(acc)   (raw v_log_f32), masked
#pragma unroll
        for (int v = 0; v < 8; ++v) {
            int M = v + hi * 8;
            float ne = fmaf(emrow[v][(size_t)s * CRF_T], LOG2E,
                            rowmaxC[M] + __builtin_amdgcn_logf(acc[v]));
            c[v] = mlds[M][s] ? ne : c[v];
        }
        // no barrier here: next iteration's barriers cover all WAR hazards
    }

    // --- denom[b] = ln2 * log2sumexp2_j(c2[b][j] + endT[j]*log2e)
    const float et2 = endT[nbase + nl] * LOG2E;
#pragma unroll
    for (int v = 0; v < 8; ++v) c[v] += et2;
    __syncthreads();   // protect rowmaxC before final reuse
#pragma unroll
    for (int v = 0; v < 8; ++v) {
        float x = c[v];
        x = fmaxf(x, __shfl_xor(x, 1, 32));
        x = fmaxf(x, __shfl_xor(x, 2, 32));
        x = fmaxf(x, __shfl_xor(x, 4, 32));
        x = fmaxf(x, __shfl_xor(x, 8, 32));
        if (nl == 0) partmax[wave][v + hi * 8] = x;
    }
    __syncthreads();
    if (tid < 16) {
        rowmaxC[tid] = fmaxf(fmaxf(partmax[0][tid], partmax[1][tid]),
                             fmaxf(partmax[2][tid], partmax[3][tid]));
    }
    __syncthreads();
#pragma unroll
    for (int v = 0; v < 8; ++v) {
        int M = v + hi * 8;
        float e = __builtin_amdgcn_exp2f(c[v] - rowmaxC[M]);
        e += __shfl_xor(e, 1, 32);
        e += __shfl_xor(e, 2, 32);
        e += __shfl_xor(e, 4, 32);
        e += __shfl_xor(e, 8, 32);
        if (nl == 0) partsum[wave][M] = e;
    }
    __syncthreads();
    if (tid < 16) {
        float sum = partsum[0][tid] + partsum[1][tid] +
                    partsum[2][tid] + partsum[3][tid];
        denom[bbase + tid] = (rowmaxC[tid] + __builtin_amdgcn_logf(sum)) * LN2;
    }
}

// ---------------------------------------------------------------------------
// Final: mean over batch of (denom - num)
// ---------------------------------------------------------------------------
__global__ void crf_reduce_kernel(const float* __restrict__ denom,
                                  const float* __restrict__ num,
                                  float* __restrict__ out) {
    __shared__ float sh[CRF_B];
    int t = threadIdx.x;
    sh[t] = denom[t] - num[t];
    __syncthreads();
    for (int off = CRF_B / 2; off > 0; off >>= 1) {
        if (t < off) sh[t] += sh[t + off];
        __syncthreads();
    }
    if (t == 0) out[0] = sh[0] * (1.0f / CRF_B);
}

extern "C" void kernel_launch(void* const* d_in, const int* in_sizes, int n_in,
                              void* d_out, int out_size, void* d_ws, size_t ws_size,
                              hipStream_t stream) {
    const float*     em     = (const float*)d_in[0];
    const long long* tags   = (const long long*)d_in[1];
    const int*       mask   = (const int*)d_in[2];
    const float*     startT = (const float*)d_in[3];
    const float*     endT   = (const float*)d_in[4];
    const float*     trans  = (const float*)d_in[5];
    float* out = (float*)d_out;

    float* denom = (float*)d_ws;           // 256 floats
    float* num   = denom + CRF_B;          // 256 floats

    crf_num_kernel<<<dim3(CRF_B / 64), dim3(64), 0, stream>>>(
        em, tags, mask, startT, endT, trans, num);
    crf_fwd_kernel<<<dim3(CRF_B / 16), dim3(128), 0, stream>>>(
        em, mask, startT, endT, trans, denom);
    crf_reduce_kernel<<<dim3(1), dim3(CRF_B), 0, stream>>>(denom, num, out);
}